// Collocation_81870666596902
// MI455X (gfx1250) — compile-verified
//
#include <hip/hip_runtime.h>

typedef float v2f __attribute__((ext_vector_type(2)));
typedef float v8f __attribute__((ext_vector_type(8)));

#define SD 4          // collocation stages
#define DD 128        // state dimension
#define PITCH 132     // LDS row pitch in floats (conflict-free for WMMA A-loads, 16B aligned)
#define NITER 24      // fixed-point sweeps (spectral radius <~0.3 -> < f32 eps)

__global__ __launch_bounds__(256, 1)
void colloc_kernel(const float* __restrict__ A_nodes,
                   const float* __restrict__ y0g,
                   const float* __restrict__ hg,
                   const float* __restrict__ ag,   // a_matrix [4][4]
                   const float* __restrict__ bg,   // b weights [4]
                   float* __restrict__ yout,       // [B][128]
                   float* __restrict__ Aout)       // pass-through copy of A_nodes
{
    __shared__ float Asm[SD * DD * PITCH]; // per-batch A, row-major, padded pitch
    __shared__ float uS[SD * DD];          // stage mix vectors u_s
    __shared__ float wS[SD * DD];          // w_s = A_s u_s
    __shared__ float rtS[SD * DD];         // r~_s = sum_j a_sj R_j
    __shared__ float y0S[DD];
    __shared__ float bRS[DD];              // sum_s b_s R_s
    __shared__ float aS[16];
    __shared__ float bvS[4];

    const int t = threadIdx.x;
    const int bidx = blockIdx.x;
    const float h = hg[bidx];

    // ---- Phase 1a: stage A_nodes[b] into LDS with gfx1250 async global->LDS DMA.
    // GVS form: vdst = per-lane LDS byte address, vaddr = per-lane 32-bit byte offset,
    // saddr = uniform 64-bit base. Tracked by ASYNCcnt (not covered by s_barrier!).
    const float* Ain = A_nodes + (size_t)bidx * (SD * DD * DD);
    #pragma unroll 4
    for (int c = 0; c < 64; ++c) {
        const int f4 = c * 256 + t;        // float4 index in [0,16384)
        const int f  = f4 << 2;
        const int s = f >> 14;
        const int i = (f >> 7) & 127;
        const int j = f & 127;
        const unsigned lds_off = (unsigned)(size_t)&Asm[(s * DD + i) * PITCH + j];
        const unsigned g_off   = (unsigned)f * 4u;
        asm volatile("global_load_async_to_lds_b128 %0, %1, %2"
                     :: "v"(lds_off), "v"(g_off), "s"(Ain) : "memory");
    }
    if (t < DD) y0S[t] = y0g[bidx * DD + t];
    if (t < 16) aS[t]  = ag[t];
    if (t < 4)  bvS[t] = bg[t];
    asm volatile("s_wait_asynccnt 0x0" ::: "memory");
    __syncthreads();

    // ---- Phase 1b: pass-through copy A to d_out, sourced from LDS (single HBM read
    // happened above); the stores drain in the background under the WMMA iterations.
    float4* Aou4 = reinterpret_cast<float4*>(Aout + (size_t)bidx * (SD * DD * DD));
    #pragma unroll 4
    for (int c = 0; c < 64; ++c) {
        const int f4 = c * 256 + t;
        const int f  = f4 << 2;
        const int s = f >> 14;
        const int i = (f >> 7) & 127;
        const int j = f & 127;
        Aou4[f4] = *reinterpret_cast<const float4*>(&Asm[(s * DD + i) * PITCH + j]);
    }

    // WMMA tiling state: wave w owns row-tile m = w (rows m*16..m*16+15) for all stages.
    const int lane = t & 31;
    const int wv   = t >> 5;                       // 0..7
    const int rowA = wv * 16 + (lane & 15);        // A-operand row for this lane
    const int kOff = (lane >> 4) << 1;             // lanes 0-15 -> K{0,1}; 16-31 -> K{2,3}

    // matvec: wS[s] = A_s @ uS[s]  for all 4 stages, via chained v_wmma_f32_16x16x4_f32.
    auto matvec = [&]() {
        v8f acc[SD];
        const v8f zero = {0.f, 0.f, 0.f, 0.f, 0.f, 0.f, 0.f, 0.f};
        #pragma unroll
        for (int s = 0; s < SD; ++s) acc[s] = zero;
        #pragma unroll 2
        for (int kc = 0; kc < 32; ++kc) {
            const int k = kc << 2;
            #pragma unroll
            for (int s = 0; s < SD; ++s) {
                // A 16x4 tile: lane holds A[row, k+kOff], A[row, k+kOff+1]  (one ds_load_b64)
                const v2f Aop = *reinterpret_cast<const v2f*>(&Asm[(s * DD + rowA) * PITCH + k + kOff]);
                // B 4x16 tile: u broadcast across all N columns; VGPR0 = K{0|2}, VGPR1 = K{1|3}
                const v2f Bop = *reinterpret_cast<const v2f*>(&uS[s * DD + k + kOff]);
                acc[s] = __builtin_amdgcn_wmma_f32_16x16x4_f32(
                    false, Aop, false, Bop, (short)0, acc[s], false, false);
            }
        }
        // Every C column equals A_s u_s; lane0 holds rows m*16+r, lane16 rows m*16+8+r.
        // Single divergent region for both writer lanes.
        if ((lane & 15) == 0) {
            const int half = (lane >> 4) << 3;     // lane0 -> 0, lane16 -> 8
            #pragma unroll
            for (int s = 0; s < SD; ++s) {
                float4 lo = {acc[s][0], acc[s][1], acc[s][2], acc[s][3]};
                float4 hi = {acc[s][4], acc[s][5], acc[s][6], acc[s][7]};
                *reinterpret_cast<float4*>(&wS[s * DD + wv * 16 + half])     = lo;
                *reinterpret_cast<float4*>(&wS[s * DD + wv * 16 + half + 4]) = hi;
            }
        }
    };

    // ---- Phase 2: R_s = A_s y0  (run matvec with u_s = y0 for all s)
    uS[t]       = y0S[t & 127];
    uS[t + 256] = y0S[t & 127];
    __syncthreads();

    // it==0: derive r~, bR, u_init from R.   1..NITER: u <- r~ + h*(a . w).
    // it==NITER+1: w holds A_s u_final -> emit y.
    for (int it = 0; it <= NITER + 1; ++it) {
        matvec();
        __syncthreads();
        const int i  = t & 127;
        const int sb = t >> 7;                    // 0 or 1
        if (it == 0) {
            const float w0 = wS[0 * DD + i], w1 = wS[1 * DD + i];
            const float w2 = wS[2 * DD + i], w3 = wS[3 * DD + i];
            #pragma unroll
            for (int q = 0; q < 2; ++q) {
                const int s = sb + 2 * q;
                const float v = aS[s * 4 + 0] * w0 + aS[s * 4 + 1] * w1 +
                                aS[s * 4 + 2] * w2 + aS[s * 4 + 3] * w3;
                rtS[s * DD + i] = v;   // r~_s
                uS[s * DD + i]  = v;   // u^(0) (corresponds to K^(0) = R)
            }
            if (sb == 0)
                bRS[i] = bvS[0] * w0 + bvS[1] * w1 + bvS[2] * w2 + bvS[3] * w3;
        } else if (it <= NITER) {
            const float w0 = wS[0 * DD + i], w1 = wS[1 * DD + i];
            const float w2 = wS[2 * DD + i], w3 = wS[3 * DD + i];
            #pragma unroll
            for (int q = 0; q < 2; ++q) {
                const int s = sb + 2 * q;
                uS[s * DD + i] = rtS[s * DD + i] +
                                 h * (aS[s * 4 + 0] * w0 + aS[s * 4 + 1] * w1 +
                                      aS[s * 4 + 2] * w2 + aS[s * 4 + 3] * w3);
            }
        } else {
            // K_s = R_s + h w_s  =>  sum_s b_s K_s = bR + h * sum_s b_s w_s
            if (t < DD) {
                const float sw = bvS[0] * wS[0 * DD + t] + bvS[1] * wS[1 * DD + t] +
                                 bvS[2] * wS[2 * DD + t] + bvS[3] * wS[3 * DD + t];
                yout[bidx * DD + t] = y0S[t] + h * bRS[t] + h * h * sw;
            }
        }
        __syncthreads();
    }
}

extern "C" void kernel_launch(void* const* d_in, const int* in_sizes, int n_in,
                              void* d_out, int out_size, void* d_ws, size_t ws_size,
                              hipStream_t stream) {
    const float* A_nodes = (const float*)d_in[0];
    const float* y0      = (const float*)d_in[1];
    const float* h       = (const float*)d_in[2];
    const float* a_mat   = (const float*)d_in[3];
    const float* b_vec   = (const float*)d_in[4];

    const int B = in_sizes[2];          // h has one entry per batch
    float* yout = (float*)d_out;        // first B*128 floats: y_next
    float* Aout = yout + (size_t)B * DD;// remainder: A_nodes pass-through

    colloc_kernel<<<B, 256, 0, stream>>>(A_nodes, y0, h, a_mat, b_vec, yout, Aout);
    (void)n_in; (void)out_size; (void)d_ws; (void)ws_size;
}